// DiscreteTokenSelection_30331059044531
// MI455X (gfx1250) — compile-verified
//
#include <hip/hip_runtime.h>
#include <stdint.h>

// ---------------------------------------------------------------------------
// Fused LayerNorm(D=4096) -> Linear(D->1) -> sigmoid(logit/T)
//   logit = rstd * (x.gw - mean*Sgw) + Sbw + b    where gw = gamma*w
// Bandwidth bound: 537MB of x at 23.3 TB/s -> ~23us floor.
// CDNA5 path: global_load_async_to_lds_b128 streaming + s_wait_asynccnt.
// ---------------------------------------------------------------------------

#define D_DIM 4096
#define WPB 8                                   // waves per block (1 row / wave)
#define CHUNK_FLOATS 1024                       // 4 KB chunk
#define NCHUNK (D_DIM / CHUNK_FLOATS)           // 4
#define INSTR_PER_CHUNK ((CHUNK_FLOATS * 4) / 512)  // 8 async b128 instrs/chunk

__device__ __forceinline__ float4 ld4(const float* p) {
  return *reinterpret_cast<const float4*>(p);
}

// ---- prep: gw[d] = gamma[d]*w[d]; scal[0]=sum(gamma*w); scal[1]=sum(beta*w)
__global__ __launch_bounds__(256) void prep_kernel(
    const float* __restrict__ gamma, const float* __restrict__ beta,
    const float* __restrict__ w, float* __restrict__ gw,
    float* __restrict__ scal) {
  __shared__ float r1[256];
  __shared__ float r2[256];
  const int tid = threadIdx.x;
  float sgw = 0.f, sbw = 0.f;
#pragma unroll
  for (int k = 0; k < D_DIM / (256 * 4); ++k) {
    const int i = (tid + k * 256) * 4;
    float4 g = ld4(gamma + i);
    float4 bb = ld4(beta + i);
    float4 ww = ld4(w + i);
    float4 gv = make_float4(g.x * ww.x, g.y * ww.y, g.z * ww.z, g.w * ww.w);
    *reinterpret_cast<float4*>(gw + i) = gv;
    sgw += (gv.x + gv.y) + (gv.z + gv.w);
    sbw += bb.x * ww.x + bb.y * ww.y + bb.z * ww.z + bb.w * ww.w;
  }
  r1[tid] = sgw;
  r2[tid] = sbw;
  __syncthreads();
  for (int s = 128; s > 0; s >>= 1) {
    if (tid < s) { r1[tid] += r1[tid + s]; r2[tid] += r2[tid + s]; }
    __syncthreads();
  }
  if (tid == 0) { scal[0] = r1[0]; scal[1] = r2[0]; }
}

// ---- main: one row per wave, async-to-LDS double-buffered streaming
template <bool USE_WS>
__global__ __launch_bounds__(256) void ln_row_kernel(
    const float* __restrict__ x, const float* __restrict__ gw,
    const float* __restrict__ scal, const float* __restrict__ gamma,
    const float* __restrict__ beta, const float* __restrict__ w,
    const float* __restrict__ bptr, const float* __restrict__ tptr,
    float* __restrict__ out, int rows) {
  __shared__ float lds[WPB * 2 * CHUNK_FLOATS];  // 8 waves * 8 KB = 64 KB
  const int lane = threadIdx.x & 31;
  const int wave = threadIdx.x >> 5;
  const int row = blockIdx.x * WPB + wave;
  if (row >= rows) return;

  float* buf0 = &lds[wave * (2 * CHUNK_FLOATS)];
  float* buf1 = buf0 + CHUNK_FLOATS;
  // low 32 bits of a generic pointer to LDS == LDS byte offset
  const uint32_t lds0 = (uint32_t)(uintptr_t)buf0 + (uint32_t)lane * 16u;
  const uint32_t lds1 = (uint32_t)(uintptr_t)buf1 + (uint32_t)lane * 16u;
  const uint64_t rowbase =
      (uint64_t)(uintptr_t)(x + (size_t)row * D_DIM) + (uint64_t)lane * 16u;

  auto issue = [&](uint32_t ldsaddr, int chunk) {
    const uint64_t g = rowbase + (uint64_t)chunk * (CHUNK_FLOATS * 4);
#pragma unroll
    for (int j = 0; j < INSTR_PER_CHUNK; ++j) {
      asm volatile("global_load_async_to_lds_b128 %0, %1, off" ::"v"(
                       ldsaddr + (uint32_t)(j * 512u)),
                   "v"(g + (uint64_t)(j * 512u))
                   : "memory");
    }
  };

  float s1 = 0.f, s2 = 0.f, s3 = 0.f, s4 = 0.f, s5 = 0.f;
  auto process = [&](const float* buf, int chunk) {
    const int cbase = chunk * CHUNK_FLOATS + lane * 4;
    const float* bc = buf + lane * 4;
#pragma unroll
    for (int k = 0; k < CHUNK_FLOATS / 128; ++k) {
      const float4 v = ld4(bc + k * 128);
      float4 gv;
      if constexpr (USE_WS) {
        gv = ld4(gw + cbase + k * 128);
      } else {
        const float4 g4 = ld4(gamma + cbase + k * 128);
        const float4 w4 = ld4(w + cbase + k * 128);
        const float4 b4 = ld4(beta + cbase + k * 128);
        gv = make_float4(g4.x * w4.x, g4.y * w4.y, g4.z * w4.z, g4.w * w4.w);
        s4 += (gv.x + gv.y) + (gv.z + gv.w);
        s5 = fmaf(b4.x, w4.x,
                  fmaf(b4.y, w4.y, fmaf(b4.z, w4.z, fmaf(b4.w, w4.w, s5))));
      }
      s1 += (v.x + v.y) + (v.z + v.w);
      s2 = fmaf(v.x, v.x, fmaf(v.y, v.y, fmaf(v.z, v.z, fmaf(v.w, v.w, s2))));
      s3 = fmaf(v.x, gv.x,
                fmaf(v.y, gv.y, fmaf(v.z, gv.z, fmaf(v.w, gv.w, s3))));
    }
  };

  issue(lds0, 0);
  issue(lds1, 1);

  asm volatile("s_wait_asynccnt 8" ::: "memory");  // chunk0 landed
  process(buf0, 0);
  asm volatile("s_wait_dscnt 0" ::: "memory");     // DS reads done before reuse
  issue(lds0, 2);

  asm volatile("s_wait_asynccnt 8" ::: "memory");  // chunk1 landed
  process(buf1, 1);
  asm volatile("s_wait_dscnt 0" ::: "memory");
  issue(lds1, 3);

  asm volatile("s_wait_asynccnt 8" ::: "memory");  // chunk2 landed
  process(buf0, 2);

  asm volatile("s_wait_asynccnt 0" ::: "memory");  // chunk3 landed
  process(buf1, 3);

#pragma unroll
  for (int off = 16; off > 0; off >>= 1) {
    s1 += __shfl_xor(s1, off, 32);
    s2 += __shfl_xor(s2, off, 32);
    s3 += __shfl_xor(s3, off, 32);
    if constexpr (!USE_WS) {
      s4 += __shfl_xor(s4, off, 32);
      s5 += __shfl_xor(s5, off, 32);
    }
  }

  if (lane == 0) {
    const float inv_d = 1.0f / (float)D_DIM;
    const float mean = s1 * inv_d;
    const float var = fmaf(-mean, mean, s2 * inv_d);
    const float rstd = rsqrtf(var + 1e-5f);
    float Sgw, Sbw;
    if constexpr (USE_WS) {
      Sgw = scal[0];
      Sbw = scal[1];
    } else {
      Sgw = s4;
      Sbw = s5;
    }
    const float logit = fmaf(rstd, s3 - mean * Sgw, Sbw + bptr[0]);
    const float z = logit / tptr[0];
    out[row] = 1.0f / (1.0f + __expf(-z));
  }
}

extern "C" void kernel_launch(void* const* d_in, const int* in_sizes, int n_in,
                              void* d_out, int out_size, void* d_ws,
                              size_t ws_size, hipStream_t stream) {
  const float* x = (const float*)d_in[0];
  const float* gamma = (const float*)d_in[1];
  const float* beta = (const float*)d_in[2];
  const float* w = (const float*)d_in[3];
  const float* b = (const float*)d_in[4];
  const float* temp = (const float*)d_in[5];
  float* out = (float*)d_out;

  const int rows = in_sizes[0] / D_DIM;
  const int blocks = (rows + WPB - 1) / WPB;

  if (ws_size >= (D_DIM + 2) * sizeof(float)) {
    float* gw = (float*)d_ws;
    float* scal = gw + D_DIM;
    prep_kernel<<<1, 256, 0, stream>>>(gamma, beta, w, gw, scal);
    ln_row_kernel<true><<<blocks, 256, 0, stream>>>(
        x, gw, scal, nullptr, nullptr, nullptr, b, temp, out, rows);
  } else {
    ln_row_kernel<false><<<blocks, 256, 0, stream>>>(
        x, nullptr, nullptr, gamma, beta, w, b, temp, out, rows);
  }
}